// ReinforceAgentQuantum_64055142253153
// MI455X (gfx1250) — compile-verified
//
#include <hip/hip_runtime.h>

// ---------------------------------------------------------------------------
// ReinforceAgentQuantum: 12-qubit statevector sim restructured as Kronecker-
// factored 64x64 complex GEMMs running on CDNA5 f32 WMMA (v_wmma_f32_16x16x4_f32),
// with batch-independent gate tables streamed into LDS via the gfx1250 async
// VMEM->LDS path (GLOBAL_LOAD_ASYNC_TO_LDS_B128 / ASYNCcnt), overlapped with
// the RX GEMMs.
//
//   state S viewed as 64x64 complex matrix (rows = wires 0..5, cols = 6..11)
//   RX layer : S <- Arx * S * Brx^T   (Arx/Brx complex, per-batch, built in LDS)
//   RY layer : S <- Ay  * S * By^T    (real, batch-independent, precomputed)
//   RZ + CZ  : S <- S .* D            (complex diagonal, precomputed)
//   out[a]   = sum_i |S_i|^2 * (1-2*bit_a(i)) * output_scaling[a]
// ---------------------------------------------------------------------------

typedef __attribute__((ext_vector_type(2))) float v2f;
typedef __attribute__((ext_vector_type(8))) float v8f;
typedef int v4i_gcc __attribute__((vector_size(16)));   // matches builtin pointee

#define QNQ   12
#define QDIM  4096
#define QNLB  15          // NL*3 blocks
#define QBLK  512         // threads per workgroup (16 waves)

// workspace layout (floats)
#define WS_AY   0
#define WS_BY   (QNLB * QDIM)
#define WS_DRE  (2 * QNLB * QDIM)
#define WS_DIM  (3 * QNLB * QDIM)

#if defined(__has_builtin)
#  if __has_builtin(__builtin_amdgcn_global_load_async_to_lds_b128) && \
      __has_builtin(__builtin_amdgcn_s_wait_asynccnt)
#    define QASYNC 1
#  endif
#endif

// 16B async global->LDS stage (falls back to a plain vector copy if the
// builtin is unavailable on this toolchain).
__device__ __forceinline__ void stage16(const float* __restrict__ g,
                                        float* __restrict__ l)
{
#ifdef QASYNC
    __builtin_amdgcn_global_load_async_to_lds_b128(
        (__attribute__((address_space(1))) v4i_gcc*)g,
        (__attribute__((address_space(3))) v4i_gcc*)l,
        0, 0);
#else
    float4 v = *(const float4*)g;
    *(float4*)l = v;
#endif
}

__device__ __forceinline__ void stage_wait()
{
#ifdef QASYNC
    __builtin_amdgcn_s_wait_asynccnt(0);
#endif
}

// ---------------------------------------------------------------------------
// 64x64 (complex) GEMM:  D = P * Q   or   D = P * Q^T   (TQ)
// PC / QC: whether P / Q have imaginary parts. One 16x16 tile per wave,
// 16 waves cover the 64x64 output. K=64 in 16 steps of 4.
// A-frag layout (ISA 7.12.2, 32-bit A 16x4): lanes 0-15 -> K {0,1},
// lanes 16-31 -> K {2,3}; M = lane&15. C/D: 8 VGPRs, M = v + 8*(lane>=16).
// ---------------------------------------------------------------------------
template <bool PC, bool QC, bool TQ>
__device__ __forceinline__ void cgemm64(float* __restrict__ dRe, float* __restrict__ dIm,
                                        const float* __restrict__ pRe, const float* __restrict__ pIm,
                                        const float* __restrict__ qRe, const float* __restrict__ qIm,
                                        int tid)
{
    const int lane = tid & 31;
    const int wave = tid >> 5;
    const int tm   = (wave >> 2) << 4;      // tile row base
    const int tn   = (wave & 3) << 4;       // tile col base
    const int m    = lane & 15;             // A-row / B-col within tile
    const int kk   = (lane >> 4) << 1;      // K sub-offset: 0 or 2
    const int pRow = (tm + m) * 64;

    v8f rr = {}; v8f ri = {}; v8f ir = {}; v8f ii = {};

    for (int k0 = 0; k0 < 64; k0 += 4) {
        const int ka = k0 + kk;
        v2f aR = {0.f, 0.f}, aI = {0.f, 0.f};
        v2f bR = {0.f, 0.f}, bI = {0.f, 0.f};

        aR[0] = pRe[pRow + ka];
        aR[1] = pRe[pRow + ka + 1];
        if (PC) { aI[0] = pIm[pRow + ka]; aI[1] = pIm[pRow + ka + 1]; }

        if (!TQ) {
            const int q0 = ka * 64 + tn + m;
            bR[0] = qRe[q0];  bR[1] = qRe[q0 + 64];
            if (QC) { bI[0] = qIm[q0]; bI[1] = qIm[q0 + 64]; }
        } else {
            const int q0 = (tn + m) * 64 + ka;
            bR[0] = qRe[q0];  bR[1] = qRe[q0 + 1];
            if (QC) { bI[0] = qIm[q0]; bI[1] = qIm[q0 + 1]; }
        }

        rr = __builtin_amdgcn_wmma_f32_16x16x4_f32(false, aR, false, bR, (short)0, rr, false, false);
        if (QC)       ri = __builtin_amdgcn_wmma_f32_16x16x4_f32(false, aR, false, bI, (short)0, ri, false, false);
        if (PC)       ir = __builtin_amdgcn_wmma_f32_16x16x4_f32(false, aI, false, bR, (short)0, ir, false, false);
        if (PC && QC) ii = __builtin_amdgcn_wmma_f32_16x16x4_f32(false, aI, false, bI, (short)0, ii, false, false);
    }

    const int mb  = (lane >> 4) << 3;       // 0 or 8
    const int col = tn + m;
#pragma unroll
    for (int v = 0; v < 8; ++v) {
        const int o = (tm + mb + v) * 64 + col;
        float re = rr[v];
        float im = 0.f;
        if (PC && QC) { re -= ii[v]; im = ri[v] + ir[v]; }
        else if (QC)  { im = ri[v]; }
        else if (PC)  { im = ir[v]; }
        dRe[o] = re;
        dIm[o] = im;
    }
}

// ---------------------------------------------------------------------------
// Precompute (batch-independent): RY Kronecker factors + fused RZ*CZ diagonals.
// One workgroup per (layer, blk).
// ---------------------------------------------------------------------------
__global__ __launch_bounds__(256)
void qprep_kernel(const float* __restrict__ weights, float* __restrict__ ws)
{
    __shared__ float yc[12], ys[12], zt[12];
    const int lb  = blockIdx.x;          // 0..14
    const int tid = threadIdx.x;

    if (tid < 12) {
        float ty = weights[lb * 24 + tid];          // RY angle, wire tid
        yc[tid] = cosf(0.5f * ty);
        ys[tid] = sinf(0.5f * ty);
        zt[tid] = weights[lb * 24 + 12 + tid];      // RZ angle, wire tid
    }
    __syncthreads();

    float* AY = ws + WS_AY  + lb * QDIM;
    float* BY = ws + WS_BY  + lb * QDIM;
    float* DR = ws + WS_DRE + lb * QDIM;
    float* DI = ws + WS_DIM + lb * QDIM;

    // Ay = kron(RY wires 0..5), By = kron(RY wires 6..11); RY = [[c,-s],[s,c]]
    for (int e = tid; e < QDIM; e += 256) {
        const int r = e >> 6, c = e & 63;
        float pa = 1.f, pb = 1.f;
#pragma unroll
        for (int j = 0; j < 6; ++j) {
            const int br = (r >> (5 - j)) & 1;
            const int bc = (c >> (5 - j)) & 1;
            pa *= (br == bc) ? yc[j]     : (br ? ys[j]     : -ys[j]);
            pb *= (br == bc) ? yc[j + 6] : (br ? ys[j + 6] : -ys[j + 6]);
        }
        AY[e] = pa;
        BY[e] = pb;
    }

    // D[i] = CZ_ring(i) * exp(i * 0.5 * sum_w (b_w ? +zt : -zt))
    for (int i = tid; i < QDIM; i += 256) {
        float ph = 0.f, cz = 1.f;
#pragma unroll
        for (int w = 0; w < 12; ++w) {
            const int b  = (i >> (11 - w)) & 1;
            const int b2 = (i >> (11 - ((w + 1) % 12))) & 1;
            ph += b ? zt[w] : -zt[w];
            if (b & b2) cz = -cz;
        }
        ph *= 0.5f;
        DR[i] = cz * cosf(ph);
        DI[i] = cz * sinf(ph);
    }
}

// ---------------------------------------------------------------------------
// Main kernel: one workgroup per batch element. Dynamic LDS = 160 KB:
//   S (re/im) 32KB, T (re/im) 32KB, F (re/im, RX factor) 32KB,
//   staged tables yA/yB/dR/dI 64KB.
// 320KB/WGP -> 2 concurrent workgroups per WGP (32 waves -> 8 waves/SIMD32).
// ---------------------------------------------------------------------------
__global__ __launch_bounds__(QBLK)
void qsim_kernel(const float* __restrict__ x,
                 const float* __restrict__ input_scaling,
                 const float* __restrict__ ws,
                 const float* __restrict__ output_scaling,
                 float* __restrict__ out)
{
    extern __shared__ float lds[];
    float* sRe = lds;
    float* sIm = lds + 4096;
    float* tRe = lds + 8192;
    float* tIm = lds + 12288;
    float* fRe = lds + 16384;
    float* fIm = lds + 20480;
    float* yA  = lds + 24576;   // staged Ay (real 64x64)
    float* yB  = lds + 28672;   // staged By
    float* dR  = lds + 32768;   // staged diag re
    float* dI  = lds + 36864;   // staged diag im
    __shared__ float cw[12], sw[12];

    const int tid = threadIdx.x;
    const int bat = blockIdx.x;
    const float* xb = x + bat * (4 * QNQ);

    // |0...0>
    for (int i = tid; i < QDIM; i += QBLK) {
        sRe[i] = (i == 0) ? 1.f : 0.f;
        sIm[i] = 0.f;
    }
    __syncthreads();

    const float* AY = ws + WS_AY;
    const float* BY = ws + WS_BY;
    const float* DR = ws + WS_DRE;
    const float* DI = ws + WS_DIM;

    for (int lb = 0; lb < QNLB; ++lb) {
        const int blk = lb % 3;

        // kick off async staging of this block's batch-independent tables;
        // the copies overlap with the RX factor build + both RX GEMMs.
        {
            const float* gy = AY + lb * QDIM;
            const float* gb = BY + lb * QDIM;
            const float* gr = DR + lb * QDIM;
            const float* gi = DI + lb * QDIM;
            const int o0 = tid * 4;            // 512 threads x 16B = 8KB/pass
            const int o1 = o0 + 2048;
            stage16(gy + o0, yA + o0);  stage16(gy + o1, yA + o1);
            stage16(gb + o0, yB + o0);  stage16(gb + o1, yB + o1);
            stage16(gr + o0, dR + o0);  stage16(gr + o1, dR + o1);
            stage16(gi + o0, dI + o0);  stage16(gi + o1, dI + o1);
        }

        // per-wire RX half-angle trig (angle = input_scaling * feature)
        if (tid < 12) {
            const float ang = input_scaling[lb * 12 + tid] * xb[(blk + 1) * QNQ + tid];
            cw[tid] = cosf(0.5f * ang);
            sw[tid] = sinf(0.5f * ang);
        }
        __syncthreads();

        // F = Arx = kron(RX wires 0..5); entry = prod(c/s) * (-i)^hamming
        for (int e = tid; e < QDIM; e += QBLK) {
            const int r = e >> 6, c = e & 63;
            float mag = 1.f;
#pragma unroll
            for (int j = 0; j < 6; ++j) {
                const int br = (r >> (5 - j)) & 1;
                const int bc = (c >> (5 - j)) & 1;
                mag *= (br == bc) ? cw[j] : sw[j];
            }
            const int h = __popc(r ^ c) & 3;
            fRe[e] = (h == 0) ? mag : ((h == 2) ? -mag : 0.f);
            fIm[e] = (h == 1) ? -mag : ((h == 3) ? mag : 0.f);
        }
        __syncthreads();

        cgemm64<true, true, false>(tRe, tIm, fRe, fIm, sRe, sIm, tid);   // T = Arx * S
        __syncthreads();

        // F = Brx = kron(RX wires 6..11)
        for (int e = tid; e < QDIM; e += QBLK) {
            const int r = e >> 6, c = e & 63;
            float mag = 1.f;
#pragma unroll
            for (int j = 0; j < 6; ++j) {
                const int br = (r >> (5 - j)) & 1;
                const int bc = (c >> (5 - j)) & 1;
                mag *= (br == bc) ? cw[6 + j] : sw[6 + j];
            }
            const int h = __popc(r ^ c) & 3;
            fRe[e] = (h == 0) ? mag : ((h == 2) ? -mag : 0.f);
            fIm[e] = (h == 1) ? -mag : ((h == 3) ? mag : 0.f);
        }
        __syncthreads();

        cgemm64<true, true, true>(sRe, sIm, tRe, tIm, fRe, fIm, tid);    // S = T * Brx^T

        stage_wait();                       // ASYNCcnt -> 0 for this wave
        __syncthreads();                    // staged LDS visible to all waves

        cgemm64<false, true, false>(tRe, tIm, yA, nullptr, sRe, sIm, tid); // T = Ay * S
        __syncthreads();

        cgemm64<true, false, true>(sRe, sIm, tRe, tIm, yB, nullptr, tid);  // S = T * By^T
        __syncthreads();

        // diagonal: RZ phases fused with CZ ring (tables already in LDS)
        for (int i = tid; i < QDIM; i += QBLK) {
            const float ar = sRe[i], ai = sIm[i];
            const float br = dR[i],  bi = dI[i];
            sRe[i] = ar * br - ai * bi;
            sIm[i] = ar * bi + ai * br;
        }
        __syncthreads();
    }

    // expvals: signed probability sums over top-4 bits
    float a0 = 0.f, a1 = 0.f, a2 = 0.f, a3 = 0.f;
    for (int i = tid; i < QDIM; i += QBLK) {
        const float p = sRe[i] * sRe[i] + sIm[i] * sIm[i];
        a0 += ((i >> 11) & 1) ? -p : p;
        a1 += ((i >> 10) & 1) ? -p : p;
        a2 += ((i >> 9) & 1) ? -p : p;
        a3 += ((i >> 8) & 1) ? -p : p;
    }
    tRe[tid]        = a0;
    tRe[512 + tid]  = a1;
    tRe[1024 + tid] = a2;
    tRe[1536 + tid] = a3;
    __syncthreads();
    for (int s = 256; s > 0; s >>= 1) {
        if (tid < s) {
            tRe[tid]        += tRe[tid + s];
            tRe[512 + tid]  += tRe[512 + tid + s];
            tRe[1024 + tid] += tRe[1024 + tid + s];
            tRe[1536 + tid] += tRe[1536 + tid + s];
        }
        __syncthreads();
    }
    if (tid < 4)
        out[bat * 4 + tid] = tRe[tid * 512] * output_scaling[tid];
}

// ---------------------------------------------------------------------------
extern "C" void kernel_launch(void* const* d_in, const int* in_sizes, int n_in,
                              void* d_out, int out_size, void* d_ws, size_t ws_size,
                              hipStream_t stream)
{
    const float* x   = (const float*)d_in[0];   // (4096, 48)
    const float* isc = (const float*)d_in[1];   // (5, 3, 12)
    const float* w   = (const float*)d_in[2];   // (5, 3, 24)
    const float* osc = (const float*)d_in[3];   // (4,)
    float* out = (float*)d_out;                 // (4096, 4)
    float* ws  = (float*)d_ws;                  // needs 60*4096 floats (~0.94 MB)

    qprep_kernel<<<QNLB, 256, 0, stream>>>(w, ws);
    qsim_kernel<<<4096, QBLK, 40960 * sizeof(float), stream>>>(x, isc, ws, osc, out);
}